// GCN_18107582120448
// MI455X (gfx1250) — compile-verified
//
#include <hip/hip_runtime.h>
#include <hip/hip_bf16.h>

typedef __attribute__((ext_vector_type(2))) float v2f;
typedef __attribute__((ext_vector_type(8))) float v8f;

// ---------------- degree / symmetric norm (with self loops) ----------------
__global__ void k_deg(const long long* __restrict__ ei, float* __restrict__ deg,
                      int E, int N) {
  int i = blockIdx.x * blockDim.x + threadIdx.x;
  if (i < E) {
    atomicAdd(&deg[(int)ei[E + i]], 1.0f);        // dst column of edge_index
  } else if (i < E + N) {
    atomicAdd(&deg[i - E], 1.0f);                 // self loop
  }
}

__global__ void k_dinv(const float* __restrict__ deg, float* __restrict__ dinv, int N) {
  int i = blockIdx.x * blockDim.x + threadIdx.x;
  if (i < N) {
    float d = deg[i];
    dinv[i] = d > 0.0f ? rsqrtf(d) : 0.0f;
  }
}

__global__ void k_norm(const long long* __restrict__ ei, const float* __restrict__ dinv,
                       float* __restrict__ norm, int E, int N) {
  int i = blockIdx.x * blockDim.x + threadIdx.x;
  if (i >= E + N) return;
  int s = (i < E) ? (int)ei[i]     : (i - E);
  int d = (i < E) ? (int)ei[E + i] : (i - E);
  norm[i] = dinv[s] * dinv[d];
}

// ---------------- dense GEMM via V_WMMA_F32_16X16X4_F32 + async LDS staging --------
// Block = 256 threads = 8 waves. Each block computes a 128-row x 16-col output slab:
// wave w -> 16x16 tile at rows [rowBlk*128 + w*16, +16), cols [col0, col0+16).
// The W column slice W[:, col0:col0+16] (DIN*16 f32, <=32KB) is staged in LDS once per
// block via global_load_async_to_lds_b32 (ASYNCcnt) so all 8 waves share it via ds_load.
//
// Fragment layouts (ISA 7.12.2, 32-bit):
//  A (16x4):  lane l<16 -> row M=l, K={k,k+1}; lanes 16..31 -> K={k+2,k+3}
//  B (4x16):  col N = lane%16; half 0 -> K={k,k+1}, half 1 -> K={k+2,k+3}
//  D (16x16): VGPR j -> row M = j + 8*(lane/16), col N = lane%16
template <int DIN>
__global__ void k_gemm_wmma(const float* __restrict__ A, const float* __restrict__ W,
                            float* __restrict__ C, int n, int Ncols) {
  __shared__ float lds_w[DIN * 16];

  const int tid  = threadIdx.x;
  const int lane = tid & 31;
  const int wave = tid >> 5;

  const int colTiles = Ncols >> 4;
  const int tileCol  = blockIdx.x % colTiles;
  const int rowBlk   = blockIdx.x / colTiles;
  const int col0     = tileCol << 4;
  const int row0     = rowBlk * 128 + wave * 16;

  // ---- cooperative async copy of W[:, col0:col0+16] into LDS ----
  // element g -> (k = g/16, c = g%16); lds_w[k*16 + c]
  for (int g = tid; g < DIN * 16; g += 256) {
    const int k = g >> 4;
    const int c = g & 15;
    const float*   gp     = W + (long long)k * Ncols + col0 + c;
    const unsigned ldsoff = (unsigned)(size_t)&lds_w[g];
    asm volatile("global_load_async_to_lds_b32 %0, %1, off"
                 :: "v"(ldsoff), "v"(gp) : "memory");
  }
  asm volatile("s_wait_asynccnt 0x0" ::: "memory");
  __syncthreads();

  const int half = lane >> 4;   // 0: lanes 0-15, 1: lanes 16-31
  const int l16  = lane & 15;

  int rowA = row0 + l16;
  if (rowA >= n) rowA = n - 1;  // clamp loads (stores guarded); keeps EXEC all-1
  const float* arow = A + (long long)rowA * DIN;

  v8f acc = {};
#pragma unroll 8
  for (int k = 0; k < DIN; k += 4) {
    const int ka = k + 2 * half;
    __builtin_prefetch(arow + k + 32, 0, 1);        // global_prefetch_b8
    v2f a;
    a.x = arow[ka];                                  // contiguous pair -> b64 load
    a.y = arow[ka + 1];
    v2f b;
    b.x = lds_w[ka * 16 + l16];                      // ds_load from staged W slice
    b.y = lds_w[(ka + 1) * 16 + l16];
    acc = __builtin_amdgcn_wmma_f32_16x16x4_f32(
        /*neg_a=*/false, a, /*neg_b=*/false, b,
        /*c_mod=*/(short)0, acc, /*reuse_a=*/false, /*reuse_b=*/false);
  }

#pragma unroll
  for (int j = 0; j < 8; ++j) {
    const int r = row0 + j + 8 * half;
    if (r < n) C[(long long)r * Ncols + col0 + l16] = acc[j];
  }
}

// ---------------- edge scatter-add: out[dst] += norm * tmp[src] ----------------
__global__ void k_scatter(const long long* __restrict__ ei, const float* __restrict__ norm,
                          const float* __restrict__ tmp, float* __restrict__ outb,
                          int E, int N, int dout) {
  const int chunks = dout >> 2;                       // float4 chunks per row
  long long gid = (long long)blockIdx.x * blockDim.x + threadIdx.x;
  const long long total = (long long)(E + N) * chunks;
  if (gid >= total) return;
  const int e = (int)(gid / chunks);
  const int c = (int)(gid % chunks) << 2;
  const int s = (e < E) ? (int)ei[e]     : (e - E);
  const int d = (e < E) ? (int)ei[E + e] : (e - E);
  const float wgt = norm[e];
  const float4 v = *(const float4*)(tmp + (long long)s * dout + c);
  float* o = outb + (long long)d * dout + c;
  atomicAdd(o + 0, wgt * v.x);
  atomicAdd(o + 1, wgt * v.y);
  atomicAdd(o + 2, wgt * v.z);
  atomicAdd(o + 3, wgt * v.w);
}

// ---------------- bias (+ optional ReLU) ----------------
__global__ void k_bias_act(float* __restrict__ h, const float* __restrict__ bias,
                           int N, int dout, int relu) {
  long long gid = (long long)blockIdx.x * blockDim.x + threadIdx.x;
  if (gid >= (long long)N * dout) return;
  const int f = (int)(gid % dout);
  float v = h[gid] + bias[f];
  h[gid] = relu ? fmaxf(v, 0.0f) : v;
}

// ---------------- global mean pool accumulation ----------------
__global__ void k_pool(const float* __restrict__ h, const long long* __restrict__ batch,
                       float* __restrict__ sums, float* __restrict__ cnt, int N) {
  long long gid = (long long)blockIdx.x * blockDim.x + threadIdx.x;
  if (gid >= (long long)N * 32) return;
  const int node = (int)(gid >> 5);
  const int f = (int)(gid & 31);
  const int g = (int)batch[node];
  atomicAdd(&sums[g * 32 + f], h[(long long)node * 32 + f]);
  if (f == 0) atomicAdd(&cnt[g], 1.0f);
}

// ---------------- tiny MLP head: relu(pooled@lw1+lb1)@lw2+lb2 ----------------
__global__ void k_head(const float* __restrict__ sums, const float* __restrict__ cnt,
                       const float* __restrict__ lw1, const float* __restrict__ lb1,
                       const float* __restrict__ lw2, const float* __restrict__ lb2,
                       float* __restrict__ out, int G) {
  const int g = blockIdx.x * blockDim.x + threadIdx.x;
  if (g >= G) return;
  const float c = fmaxf(cnt[g], 1.0f);
  float pooled[32];
#pragma unroll
  for (int f = 0; f < 32; ++f) pooled[f] = sums[g * 32 + f] / c;
  float z[16];
#pragma unroll
  for (int j = 0; j < 16; ++j) {
    float s = lb1[j];
#pragma unroll
    for (int f = 0; f < 32; ++f) s += pooled[f] * lw1[f * 16 + j];
    z[j] = fmaxf(s, 0.0f);
  }
#pragma unroll
  for (int o = 0; o < 2; ++o) {
    float s = lb2[o];
#pragma unroll
    for (int j = 0; j < 16; ++j) s += z[j] * lw2[j * 2 + o];
    out[g * 2 + o] = s;
  }
}

static inline void launch_gemm(int din, const float* A, const float* W, float* C,
                               int n, int dout, hipStream_t stream) {
  const int colTiles = dout >> 4;
  const int rowBlks  = (n + 127) / 128;
  const unsigned blocks = (unsigned)(rowBlks * colTiles);
  switch (din) {
    case 512: k_gemm_wmma<512><<<blocks, 256, 0, stream>>>(A, W, C, n, dout); break;
    case 256: k_gemm_wmma<256><<<blocks, 256, 0, stream>>>(A, W, C, n, dout); break;
    case 128: k_gemm_wmma<128><<<blocks, 256, 0, stream>>>(A, W, C, n, dout); break;
    default:  k_gemm_wmma<64> <<<blocks, 256, 0, stream>>>(A, W, C, n, dout); break;
  }
}

extern "C" void kernel_launch(void* const* d_in, const int* in_sizes, int n_in,
                              void* d_out, int out_size, void* d_ws, size_t ws_size,
                              hipStream_t stream) {
  const float*     x     = (const float*)d_in[0];
  const long long* ei    = (const long long*)d_in[1];   // int64 [2, E]
  const long long* batch = (const long long*)d_in[2];   // int64 [N]
  const float* w[4] = {(const float*)d_in[3], (const float*)d_in[5],
                       (const float*)d_in[7], (const float*)d_in[9]};
  const float* b[4] = {(const float*)d_in[4], (const float*)d_in[6],
                       (const float*)d_in[8], (const float*)d_in[10]};
  const float* lw1 = (const float*)d_in[11];
  const float* lb1 = (const float*)d_in[12];
  const float* lw2 = (const float*)d_in[13];
  const float* lb2 = (const float*)d_in[14];
  float* out = (float*)d_out;

  const int N = in_sizes[0] / 512;
  const int E = in_sizes[1] / 2;
  const int G = out_size / 2;

  // workspace layout (floats)
  float* ws   = (float*)d_ws;
  float* deg  = ws;                                  // N
  float* dinv = deg + N;                             // N
  float* norm = dinv + N;                            // E + N
  float* bufA = norm + (size_t)(E + N);              // N * 256
  float* bufB = bufA + (size_t)N * 256;              // N * 256
  float* bufC = bufB + (size_t)N * 256;              // N * 128
  float* sums = bufC + (size_t)N * 128;              // G * 32
  float* cnt  = sums + (size_t)G * 32;               // G

  // ---- normalization coefficients ----
  (void)hipMemsetAsync(deg, 0, (size_t)N * sizeof(float), stream);
  {
    const int tot = E + N;
    k_deg <<<(tot + 255) / 256, 256, 0, stream>>>(ei, deg, E, N);
    k_dinv<<<(N   + 255) / 256, 256, 0, stream>>>(deg, dinv, N);
    k_norm<<<(tot + 255) / 256, 256, 0, stream>>>(ei, dinv, norm, E, N);
  }

  // ---- 4 GCN layers ----
  const int dins[4]  = {512, 256, 128, 64};
  const int douts[4] = {256, 128, 64, 32};
  const float* hin[4]  = {x,    bufB, bufA, bufB};
  float*       tmps[4] = {bufA, bufC, bufC, bufC};
  float*       houts[4]= {bufB, bufA, bufB, bufA};

  for (int L = 0; L < 4; ++L) {
    const int din = dins[L], dout = douts[L];

    // tmp = hin @ W  (wmma f32, LDS-staged weights)
    launch_gemm(din, hin[L], w[L], tmps[L], N, dout, stream);

    // hout = scatter-add(norm * tmp[src]) ; then + bias (+ relu for L<3)
    (void)hipMemsetAsync(houts[L], 0, (size_t)N * dout * sizeof(float), stream);
    const long long total = (long long)(E + N) * (dout / 4);
    k_scatter<<<(unsigned)((total + 255) / 256), 256, 0, stream>>>(ei, norm, tmps[L],
                                                                   houts[L], E, N, dout);
    const long long nel = (long long)N * dout;
    k_bias_act<<<(unsigned)((nel + 255) / 256), 256, 0, stream>>>(houts[L], b[L], N, dout,
                                                                  (L < 3) ? 1 : 0);
  }

  // ---- global mean pool + MLP head ----
  (void)hipMemsetAsync(sums, 0, (size_t)G * 32 * sizeof(float), stream);
  (void)hipMemsetAsync(cnt,  0, (size_t)G * sizeof(float), stream);
  const long long pt = (long long)N * 32;
  k_pool<<<(unsigned)((pt + 255) / 256), 256, 0, stream>>>(houts[3], batch, sums, cnt, N);
  k_head<<<1, (G + 31) / 32 * 32, 0, stream>>>(sums, cnt, lw1, lb1, lw2, lb2, out, G);
}